// SSAtt_cg_42202348650944
// MI455X (gfx1250) — compile-verified
//
#include <hip/hip_runtime.h>
#include <hip/hip_bf16.h>

// ---------------------------------------------------------------------------
// MI455X (gfx1250) implementation of the SPD submanifold-attention network.
//
//   * All congruence transforms Y = W^T X W (incl. fused sub-block gathers)
//     run through V_WMMA_F32_16X16X4_F32 with LDS-staged, zero-padded panels:
//     the K-loop is branch-free (ds_load_b64 x2 + wmma, unrolled x4).
//   * Batched symmetric eigh via cyclic Jacobi in LDS (one matrix / WG),
//     with spectral function (rect / log / exp-rect) + U f(S) U^T rebuild.
//   * Attention simplified exactly: softmax(.., axis=-2)[:, :, -1] only needs
//     the LAST Q submanifold; exp-then-rectify fused into one eigh.
// ---------------------------------------------------------------------------

typedef __attribute__((ext_vector_type(2))) float v2f;
typedef __attribute__((ext_vector_type(8))) float v8f;

#define EPS_JIT_F   0.001f
#define EPS_RECT_F  0.0001f
#define JACOBI_SWEEPS 8

__device__ __forceinline__ v8f wmma_f32_16x16x4(v2f a, v2f b, v8f c) {
  // D = A(16x4, f32) * B(4x16, f32) + C(16x16, f32)
  return __builtin_amdgcn_wmma_f32_16x16x4_f32(
      /*neg_a=*/false, a, /*neg_b=*/false, b,
      /*c_mod=*/(short)0, c, /*reuse_a=*/false, /*reuse_b=*/false);
}

// ---------------------------------------------------------------------------
// Generic batched GEMM: C[bi] = op(A[bi]) * B[bi]
//   mode 0: A is M x K (row-major) if transA==0, else stored K x M (read ^T)
//   mode 2: A is a gathered sub-block of a d^2 x d^2 parent:
//             A[a][c] = X[row(a)][row(c)], row(e) = (gi + e/k)*d + (gj + e%k)
//   C addressing: Cb = C + pb*cPar + (cOff + sub)*cSub  with bi = pb*NS + sub
//
// Block = NW waves (NW <= 4). All waves share one 16-row A panel (tile row
// blockIdx.y); wave w computes output tile column (blockIdx.z*NW + w).
// A/B are staged into LDS zero-padded to Kp = ceil16(K): all bounds / gather /
// transpose logic is in the staging phase; the K-loop is branch-free.
// B panel is pair-interleaved so each lane's (k,k+1) fragment is one b64 load.
// ---------------------------------------------------------------------------
#define GEMM_KP_MAX 112
#define GEMM_NW_MAX 4

__global__ void k_gemm_wmma(const float* __restrict__ A, long long sA, int transA,
                            const float* __restrict__ B, long long sB,
                            float* __restrict__ C, long long cPar, long long cSub, int cOff,
                            int M, int N, int K, int NS,
                            int mode, int d, int ksub, int ns)
{
  __shared__ float Alds[16 * GEMM_KP_MAX];                 // [mi][k]
  __shared__ float Blds[GEMM_NW_MAX * 16 * GEMM_KP_MAX];   // [w][(k>>1)][n][k&1]

  int bi  = blockIdx.x;
  int pb  = bi / NS;
  int sub = bi - pb * NS;
  int gi = 0, gj = 0, npar = 0;
  const float* Ab;
  if (mode == 2) {
    gi = sub / ns; gj = sub - gi * ns;
    npar = d * d;
    Ab = A + (long long)pb * sA;
  } else {
    Ab = A + (long long)bi * sA;
  }
  const float* Bb = B + (long long)bi * sB;
  float* Cb = C + (long long)pb * cPar + (long long)(cOff + sub) * cSub;

  int NT   = blockDim.x;
  int tid  = threadIdx.x;
  int wid  = tid >> 5;
  int NW   = NT >> 5;
  int lane = tid & 31;
  int half = lane >> 4;      // 0 | 1
  int lid  = lane & 15;
  int koff = half << 1;      // 0 | 2

  int Kp = (K + 15) & ~15;
  int tm = blockIdx.y << 4;
  int tnBase = blockIdx.z * NW;  // in 16-wide tiles

  // ---- zero pads ----
  for (int e = tid; e < 16 * Kp; e += NT) Alds[e] = 0.f;
  for (int e = tid; e < NW * 16 * Kp; e += NT) Blds[e] = 0.f;
  __syncthreads();

  // ---- stage A panel: rows [tm, tm+16) x [0, K) ----
  for (int e = tid; e < 16 * K; e += NT) {
    int mi = e / K, k = e - mi * K;
    int am = tm + mi;
    float v = 0.f;
    if (am < M) {
      if (mode == 2) {
        int r = (gi + am / ksub) * d + (gj + am % ksub);
        int c = (gi + k  / ksub) * d + (gj + k  % ksub);
        v = Ab[(long long)r * npar + c];
      } else if (transA) {
        v = Ab[(long long)k * M + am];
      } else {
        v = Ab[(long long)am * K + k];
      }
    }
    Alds[mi * Kp + k] = v;
  }
  // ---- stage B panels, pair-interleaved ----
  for (int e = tid; e < NW * 16 * K; e += NT) {
    int w   = e / (16 * K);
    int rem = e - w * 16 * K;
    int k   = rem >> 4;
    int ni  = rem & 15;
    int bn  = ((tnBase + w) << 4) + ni;
    float v = (bn < N) ? Bb[(long long)k * N + bn] : 0.f;
    Blds[w * 16 * Kp + ((k >> 1) << 5) + (ni << 1) + (k & 1)] = v;
  }
  __syncthreads();

  const float* Ap = Alds;
  const float* Bp = Blds + wid * 16 * Kp;

  // ---- branch-free K loop: 2x ds_load_b64 + wmma, unrolled x4 ----
  v8f acc = {0.f, 0.f, 0.f, 0.f, 0.f, 0.f, 0.f, 0.f};
  for (int k0 = 0; k0 < Kp; k0 += 16) {
#pragma unroll
    for (int u = 0; u < 4; ++u) {
      int kk = k0 + (u << 2) + koff;               // even
      v2f a = *(const v2f*)(Ap + lid * Kp + kk);
      v2f b = *(const v2f*)(Bp + ((kk >> 1) << 5) + (lid << 1));
      acc = wmma_f32_16x16x4(a, b, acc);
    }
  }

  int tn = (tnBase + wid) << 4;
#pragma unroll
  for (int v = 0; v < 8; ++v) {
    int m = tm + v + (half << 3);
    int n = tn + lid;
    if (m < M && n < N) Cb[(long long)m * N + n] = acc[v];
  }
}

// ---------------------------------------------------------------------------
// Batched symmetric eigh (cyclic Jacobi) + spectral function + rebuild.
//   func 0: f(s) = max(s, EPS_RECT)                (spd_rectified)
//   func 1: f(s) = log(max(s, 1e-8)), diag jit     (tensor_log)
//   func 2: f(s) = max(exp(s), EPS_RECT)           (tensor_exp o rectify, fused)
// In-place safe (input fully staged to LDS before output is written).
// ---------------------------------------------------------------------------
#define EIG_MAXN 81
#define EIG_LD   82

__global__ void k_eigh_apply(const float* __restrict__ in, float* __restrict__ out,
                             int n, int count, int func, float jit)
{
  __shared__ float Am[EIG_MAXN * EIG_LD];
  __shared__ float Um[EIG_MAXN * EIG_LD];
  __shared__ float fs[EIG_MAXN];

  int b = blockIdx.x;
  if (b >= count) return;
  const float* src = in + (long long)b * n * n;
  int t = threadIdx.x, NT = blockDim.x;

  for (int e = t; e < n * n; e += NT) {
    int i = e / n, j = e - i * n;
    float v = 0.5f * (src[(long long)i * n + j] + src[(long long)j * n + i]); // _sym
    if (i == j && n > 1) v += jit * (1.0f + (float)i / (float)(n - 1));       // diag jitter
    Am[i * EIG_LD + j] = v;
    Um[i * EIG_LD + j] = (i == j) ? 1.0f : 0.0f;
  }
  __syncthreads();

  for (int sw = 0; sw < JACOBI_SWEEPS; ++sw) {
    for (int p = 0; p < n - 1; ++p) {
      for (int q = p + 1; q < n; ++q) {
        // all threads compute the identical rotation from LDS
        float apq = Am[p * EIG_LD + q];
        float app = Am[p * EIG_LD + p];
        float aqq = Am[q * EIG_LD + q];
        float c, s;
        if (fabsf(apq) < 1e-12f) { c = 1.0f; s = 0.0f; }
        else {
          float theta = 0.5f * (aqq - app) / apq;
          float tt = 1.0f / (fabsf(theta) + sqrtf(theta * theta + 1.0f));
          if (theta < 0.0f) tt = -tt;
          c = rsqrtf(tt * tt + 1.0f);
          s = tt * c;
        }
        __syncthreads();            // everyone has read app/apq/aqq
        // column rotation: A <- A J, U <- U J
        for (int kk = t; kk < n; kk += NT) {
          float akp = Am[kk * EIG_LD + p], akq = Am[kk * EIG_LD + q];
          Am[kk * EIG_LD + p] = c * akp - s * akq;
          Am[kk * EIG_LD + q] = s * akp + c * akq;
          float ukp = Um[kk * EIG_LD + p], ukq = Um[kk * EIG_LD + q];
          Um[kk * EIG_LD + p] = c * ukp - s * ukq;
          Um[kk * EIG_LD + q] = s * ukp + c * ukq;
        }
        __syncthreads();
        // row rotation: A <- J^T A
        for (int kk = t; kk < n; kk += NT) {
          float apk = Am[p * EIG_LD + kk], aqk = Am[q * EIG_LD + kk];
          Am[p * EIG_LD + kk] = c * apk - s * aqk;
          Am[q * EIG_LD + kk] = s * apk + c * aqk;
        }
        __syncthreads();
      }
    }
  }

  for (int kk = t; kk < n; kk += NT) {
    float sv = Am[kk * EIG_LD + kk];
    float fv;
    if (func == 0)      fv = fmaxf(sv, EPS_RECT_F);
    else if (func == 1) fv = logf(fmaxf(sv, 1e-8f));
    else                fv = fmaxf(expf(sv), EPS_RECT_F);
    fs[kk] = fv;
  }
  __syncthreads();

  float* dst = out + (long long)b * n * n;
  for (int e = t; e < n * n; e += NT) {
    int i = e / n, j = e - i * n;
    float acc = 0.f;
    for (int kk = 0; kk < n; ++kk)
      acc += Um[i * EIG_LD + kk] * fs[kk] * Um[j * EIG_LD + kk];
    dst[e] = acc;
  }
}

// ---------------------------------------------------------------------------
// Attention weights: only the LAST column of the softmax matters.
//   score_i = 1 / (1 + log1p(||LK_i - LQ_last||^2)), softmax over i.
// blockDim must be 64; nv <= 64.
// ---------------------------------------------------------------------------
__global__ void k_attn_weights(const float* __restrict__ LK,     // (B, nv, 16)
                               const float* __restrict__ LQlast, // (B, 16)
                               float* __restrict__ w,            // (B, nv)
                               int nv)
{
  __shared__ float sc[64];
  int b = blockIdx.x;
  int t = threadIdx.x;
  float myscore = 0.f;
  if (t < nv) {
    const float* kp = LK + ((long long)b * nv + t) * 16;
    const float* qp = LQlast + (long long)b * 16;
    float e = 0.f;
    for (int i = 0; i < 16; ++i) { float d = kp[i] - qp[i]; e += d * d; }
    myscore = 1.0f / (1.0f + log1pf(e));
  }
  sc[t] = (t < nv) ? myscore : -1e30f;
  __syncthreads();
  for (int o = 32; o > 0; o >>= 1) { if (t < o) sc[t] = fmaxf(sc[t], sc[t + o]); __syncthreads(); }
  float mx = sc[0];
  __syncthreads();
  float ex = (t < nv) ? __expf(myscore - mx) : 0.f;
  sc[t] = ex;
  __syncthreads();
  for (int o = 32; o > 0; o >>= 1) { if (t < o) sc[t] += sc[t + o]; __syncthreads(); }
  float denom = sc[0];
  if (t < nv) w[(long long)b * nv + t] = ex / denom;
}

// mean_log[b] = sum_i w[b,i] * LV[b,i,:,:]
__global__ void k_wsum(const float* __restrict__ LV, const float* __restrict__ w,
                       float* __restrict__ out, int nv, int nn)
{
  int b = blockIdx.x;
  for (int e = threadIdx.x; e < nn; e += blockDim.x) {
    float acc = 0.f;
    for (int i = 0; i < nv; ++i)
      acc += w[(long long)b * nv + i] * LV[((long long)b * nv + i) * nn + e];
    out[(long long)b * nn + e] = acc;
  }
}

// feat = triu(log(x3)) (325), out = feat @ lin_w^T + lin_b   -> (B, 9)
__global__ void k_final(const float* __restrict__ Llog,  // (B, 25, 25)
                        const float* __restrict__ lw,    // (9, 325)
                        const float* __restrict__ lb,    // (9)
                        float* __restrict__ out)         // (B, 9)
{
  int b = blockIdx.x;
  int t = threadIdx.x;
  if (t < 9) {
    const float* L = Llog + (long long)b * 625;
    float acc = lb[t];
    int f = 0;
    for (int i = 0; i < 25; ++i)
      for (int j = i; j < 25; ++j)
        acc += L[i * 25 + j] * lw[t * 325 + (f++)];
    out[b * 9 + t] = acc;
  }
}

// ---------------------------------------------------------------------------
// Host orchestration
// ---------------------------------------------------------------------------
static void gemm(hipStream_t st, const float* A, long long sA, int transA,
                 const float* B, long long sB,
                 float* C, long long cPar, long long cSub, int cOff,
                 int M, int N, int K, int nbatch, int NS,
                 int mode, int d, int ksub, int ns)
{
  int NW = (N + 15) / 16;
  if (NW > GEMM_NW_MAX) NW = GEMM_NW_MAX;
  dim3 g((unsigned)nbatch, (unsigned)((M + 15) / 16),
         (unsigned)((N + 16 * NW - 1) / (16 * NW)));
  k_gemm_wmma<<<g, dim3(32 * NW), 0, st>>>(A, sA, transA, B, sB, C, cPar, cSub, cOff,
                                           M, N, K, NS, mode, d, ksub, ns);
}

extern "C" void kernel_launch(void* const* d_in, const int* in_sizes, int n_in,
                              void* d_out, int out_size, void* d_ws, size_t ws_size,
                              hipStream_t stream)
{
  (void)in_sizes; (void)n_in; (void)out_size; (void)ws_size;
  const int B = 512;

  const float* x     = (const float*)d_in[0];   // (512,100,100)
  const float* w_t1  = (const float*)d_in[1];   // (100,81)
  const float* w_t2[3] = {(const float*)d_in[2], (const float*)d_in[3], (const float*)d_in[4]};   // (81,49)
  const float* w_t3[4] = {(const float*)d_in[5], (const float*)d_in[6], (const float*)d_in[7], (const float*)d_in[8]}; // (49,25)
  const float* q1[3] = {(const float*)d_in[9],  (const float*)d_in[10], (const float*)d_in[11]};  // (k2,4)
  const float* k1[3] = {(const float*)d_in[12], (const float*)d_in[13], (const float*)d_in[14]};
  const float* v1[3] = {(const float*)d_in[15], (const float*)d_in[16], (const float*)d_in[17]};  // (k2,49)
  const float* q2[4] = {(const float*)d_in[18], (const float*)d_in[19], (const float*)d_in[20], (const float*)d_in[21]};
  const float* k2w[4] = {(const float*)d_in[22], (const float*)d_in[23], (const float*)d_in[24], (const float*)d_in[25]};
  const float* v2w[4] = {(const float*)d_in[26], (const float*)d_in[27], (const float*)d_in[28], (const float*)d_in[29]};
  const float* lin_w = (const float*)d_in[30];  // (9,325)
  const float* lin_b = (const float*)d_in[31];  // (9)
  float* out = (float*)d_out;

  float* ws = (float*)d_ws;
  size_t off = 0;
  auto alloc = [&](size_t nflt) { float* p = ws + off; off += nflt; return p; };

  float* TBUF = alloc((size_t)B * 100 * 81);          // shared GEMM intermediate (max use)
  float* x1   = alloc((size_t)B * 81 * 81);
  float* x2s  = alloc((size_t)3 * B * 49 * 49);       // x21,x22,x23
  float* LV1  = alloc((size_t)B * 41 * 49 * 49);
  float* LK1  = alloc((size_t)B * 41 * 16);
  float* LQ1  = alloc((size_t)B * 16);
  float* w1   = alloc((size_t)B * 41);
  float* ml1  = alloc((size_t)B * 49 * 49);           // mean_log1 -> x2 (in place)
  float* x3s  = alloc((size_t)4 * B * 25 * 25);       // x31..x34
  float* LV2  = alloc((size_t)B * 30 * 625);
  float* LK2  = alloc((size_t)B * 30 * 16);
  float* LQ2  = alloc((size_t)B * 16);
  float* w2   = alloc((size_t)B * 30);
  float* ml2  = alloc((size_t)B * 625);               // mean_log2 -> x3 (in place)
  float* lx3  = alloc((size_t)B * 625);

  // ---- stage 1: x1 = rect(Wt1^T x Wt1), 100 -> 81 ----
  gemm(stream, x, 10000, 0, w_t1, 0, TBUF, 8100, 0, 0, 100, 81, 100, B, 1, 0, 0, 0, 0);
  gemm(stream, w_t1, 0, 1, TBUF, 8100, x1, 6561, 0, 0, 81, 81, 100, B, 1, 0, 0, 0, 0);
  k_eigh_apply<<<B, 96, 0, stream>>>(x1, x1, 81, B, 0, 0.f);

  // ---- stage 2: x2i = rect(Wt2i^T x1 Wt2i), 81 -> 49 ----
  for (int i = 0; i < 3; ++i) {
    float* xi = x2s + (size_t)i * B * 2401;
    gemm(stream, x1, 6561, 0, w_t2[i], 0, TBUF, 3969, 0, 0, 81, 49, 81, B, 1, 0, 0, 0, 0);
    gemm(stream, w_t2[i], 0, 1, TBUF, 3969, xi, 2401, 0, 0, 49, 49, 81, B, 1, 0, 0, 0, 0);
  }
  k_eigh_apply<<<3 * B, 64, 0, stream>>>(x2s, x2s, 49, 3 * B, 0, 0.f);

  // ---- stage 3: attention 1 (d=7; groups k=2/6/7 on x21/x22/x23) ----
  {
    const int ks[3] = {2, 6, 7}, nss[3] = {6, 2, 1}, goff[3] = {0, 36, 40};
    for (int g = 0; g < 3; ++g) {
      int k = ks[g], ns = nss[g], NS = ns * ns, k2 = k * k;
      const float* src = x2s + (size_t)g * B * 2401;
      // LV slice: (W_v^T S_sub W_v) -> 49x49
      gemm(stream, src, 2401, 0, v1[g], 0, TBUF, (long long)NS * k2 * 49, (long long)k2 * 49, 0,
           k2, 49, k2, B * NS, NS, 2, 7, k, ns);
      gemm(stream, v1[g], 0, 1, TBUF, (long long)k2 * 49, LV1, 41LL * 2401, 2401, goff[g],
           49, 49, k2, B * NS, NS, 0, 0, 0, 0);
      // LK slice: -> 4x4
      gemm(stream, src, 2401, 0, k1[g], 0, TBUF, (long long)NS * k2 * 4, (long long)k2 * 4, 0,
           k2, 4, k2, B * NS, NS, 2, 7, k, ns);
      gemm(stream, k1[g], 0, 1, TBUF, (long long)k2 * 4, LK1, 41LL * 16, 16, goff[g],
           4, 4, k2, B * NS, NS, 0, 0, 0, 0);
    }
    // LQ: only the last submanifold (full 49x49 block of x23) is needed
    const float* src2 = x2s + (size_t)2 * B * 2401;
    gemm(stream, src2, 2401, 0, q1[2], 0, TBUF, 196, 0, 0, 49, 4, 49, B, 1, 0, 0, 0, 0);
    gemm(stream, q1[2], 0, 1, TBUF, 196, LQ1, 16, 0, 0, 4, 4, 49, B, 1, 0, 0, 0, 0);

    k_eigh_apply<<<B * 41, 64, 0, stream>>>(LV1, LV1, 49, B * 41, 1, EPS_JIT_F); // tensor_log
    k_eigh_apply<<<B * 41, 32, 0, stream>>>(LK1, LK1, 4, B * 41, 1, EPS_JIT_F);
    k_eigh_apply<<<B, 32, 0, stream>>>(LQ1, LQ1, 4, B, 1, EPS_JIT_F);

    k_attn_weights<<<B, 64, 0, stream>>>(LK1, LQ1, w1, 41);
    k_wsum<<<B, 128, 0, stream>>>(LV1, w1, ml1, 41, 2401);
    k_eigh_apply<<<B, 64, 0, stream>>>(ml1, ml1, 49, B, 2, 0.f); // exp + rectify fused -> x2
  }

  // ---- stage 4: x3i = rect(Wt3i^T x2 Wt3i), 49 -> 25 ----
  for (int i = 0; i < 4; ++i) {
    float* xi = x3s + (size_t)i * B * 625;
    gemm(stream, ml1, 2401, 0, w_t3[i], 0, TBUF, 1225, 0, 0, 49, 25, 49, B, 1, 0, 0, 0, 0);
    gemm(stream, w_t3[i], 0, 1, TBUF, 1225, xi, 625, 0, 0, 25, 25, 49, B, 1, 0, 0, 0, 0);
  }
  k_eigh_apply<<<4 * B, 32, 0, stream>>>(x3s, x3s, 25, 4 * B, 0, 0.f);

  // ---- stage 5: attention 2 (d=5; groups k=2/3/4/5 on x31..x34) ----
  {
    const int ks[4] = {2, 3, 4, 5}, nss[4] = {4, 3, 2, 1}, goff[4] = {0, 16, 25, 29};
    for (int g = 0; g < 4; ++g) {
      int k = ks[g], ns = nss[g], NS = ns * ns, k2 = k * k;
      const float* src = x3s + (size_t)g * B * 625;
      gemm(stream, src, 625, 0, v2w[g], 0, TBUF, (long long)NS * k2 * 25, (long long)k2 * 25, 0,
           k2, 25, k2, B * NS, NS, 2, 5, k, ns);
      gemm(stream, v2w[g], 0, 1, TBUF, (long long)k2 * 25, LV2, 30LL * 625, 625, goff[g],
           25, 25, k2, B * NS, NS, 0, 0, 0, 0);
      gemm(stream, src, 625, 0, k2w[g], 0, TBUF, (long long)NS * k2 * 4, (long long)k2 * 4, 0,
           k2, 4, k2, B * NS, NS, 2, 5, k, ns);
      gemm(stream, k2w[g], 0, 1, TBUF, (long long)k2 * 4, LK2, 30LL * 16, 16, goff[g],
           4, 4, k2, B * NS, NS, 0, 0, 0, 0);
    }
    const float* src3 = x3s + (size_t)3 * B * 625;
    gemm(stream, src3, 625, 0, q2[3], 0, TBUF, 100, 0, 0, 25, 4, 25, B, 1, 0, 0, 0, 0);
    gemm(stream, q2[3], 0, 1, TBUF, 100, LQ2, 16, 0, 0, 4, 4, 25, B, 1, 0, 0, 0, 0);

    k_eigh_apply<<<B * 30, 32, 0, stream>>>(LV2, LV2, 25, B * 30, 1, EPS_JIT_F);
    k_eigh_apply<<<B * 30, 32, 0, stream>>>(LK2, LK2, 4, B * 30, 1, EPS_JIT_F);
    k_eigh_apply<<<B, 32, 0, stream>>>(LQ2, LQ2, 4, B, 1, EPS_JIT_F);

    k_attn_weights<<<B, 64, 0, stream>>>(LK2, LQ2, w2, 30);
    k_wsum<<<B, 128, 0, stream>>>(LV2, w2, ml2, 30, 625);
    k_eigh_apply<<<B, 32, 0, stream>>>(ml2, ml2, 25, B, 2, 0.f); // -> x3
  }

  // ---- final: feat = triu(tensor_log(x3)); out = feat @ lin_w^T + lin_b ----
  k_eigh_apply<<<B, 32, 0, stream>>>(ml2, lx3, 25, B, 1, EPS_JIT_F);
  k_final<<<B, 32, 0, stream>>>(lx3, lin_w, lin_b, out);
}